// SelfAttentionPool_34522947125344
// MI455X (gfx1250) — compile-verified
//
#include <hip/hip_runtime.h>
#include <hip/hip_bf16.h>
#include <math.h>

typedef __attribute__((ext_vector_type(2))) float v2f;
typedef __attribute__((ext_vector_type(8))) float v8f;

#define FDIM 128
#define NPG 100
#define KNUM 80

// ---------------------------------------------------------------------------
// Kernel 1: y = X @ theta  via V_WMMA_F32_16X16X4_F32 (exact f32 matmul)
// 128 threads = 4 waves; each wave owns 16 rows; tile = 64 rows x 128 cols.
// ---------------------------------------------------------------------------
#define TILE_ROWS 64
#define LDS_STRIDE 132   // pad: row-to-row bank advance = 4 -> conflict-free b64 reads

__global__ __launch_bounds__(128) void yproj_wmma_kernel(
    const float* __restrict__ x, const float* __restrict__ theta,
    float* __restrict__ y, int N) {
  __shared__ float xs[TILE_ROWS * LDS_STRIDE];
  __shared__ float th[FDIM];

  const int t = threadIdx.x;
  const int row0 = blockIdx.x * TILE_ROWS;

  if (t < FDIM) th[t] = theta[t];

  // Cooperative coalesced tile load: 64 rows * 32 float4 = 2048 float4
#pragma unroll
  for (int i = 0; i < 16; ++i) {
    int idx = t + i * 128;      // 0..2047
    int r = idx >> 5;           // row in tile
    int c4 = idx & 31;          // float4 within row
    float4 v = make_float4(0.f, 0.f, 0.f, 0.f);
    int gr = row0 + r;
    if (gr < N) v = *(const float4*)(x + (size_t)gr * FDIM + c4 * 4);
    *(float4*)(xs + r * LDS_STRIDE + c4 * 4) = v;
  }
  __syncthreads();

  const int wave = t >> 5;
  const int lane = t & 31;
  const int m = lane & 15;
  const int kh = (lane >> 4) << 1;      // 0 for lanes 0-15, 2 for lanes 16-31
  const float* arow = xs + (wave * 16 + m) * LDS_STRIDE + kh;

  v8f acc = {0.f, 0.f, 0.f, 0.f, 0.f, 0.f, 0.f, 0.f};
#pragma unroll 4
  for (int kc = 0; kc < FDIM / 4; ++kc) {
    const int kb = kc * 4;
    v2f a, b;
    a.x = arow[kb];
    a.y = arow[kb + 1];
    // broadcast theta into all 16 B columns -> every D column equals y-tile
    b.x = th[kb + kh];
    b.y = th[kb + kh + 1];
    acc = __builtin_amdgcn_wmma_f32_16x16x4_f32(
        false, a, false, b, (short)0, acc, false, false);
  }

  // D layout: vgpr r, lane 0 -> (M=r, N=0); lane 16 -> (M=r+8, N=0)
  const int gbase = row0 + wave * 16;
  if (gbase + 16 <= N) {
    // fast path: two b128 stores per contributing lane
    if (lane == 0) {
      *(float4*)(y + gbase)     = make_float4(acc[0], acc[1], acc[2], acc[3]);
      *(float4*)(y + gbase + 4) = make_float4(acc[4], acc[5], acc[6], acc[7]);
    } else if (lane == 16) {
      *(float4*)(y + gbase + 8)  = make_float4(acc[0], acc[1], acc[2], acc[3]);
      *(float4*)(y + gbase + 12) = make_float4(acc[4], acc[5], acc[6], acc[7]);
    }
  } else {
    if (lane == 0) {
#pragma unroll
      for (int r = 0; r < 8; ++r) {
        int gr = gbase + r;
        if (gr < N) y[gr] = acc[r];
      }
    } else if (lane == 16) {
#pragma unroll
      for (int r = 0; r < 8; ++r) {
        int gr = gbase + 8 + r;
        if (gr < N) y[gr] = acc[r];
      }
    }
  }
}

// ---------------------------------------------------------------------------
// Kernel 2: one workgroup per graph. deg/z in LDS (ds_add_f32), stable top-80,
// gather+scale x rows, relabel edges, batch ids. Outputs stored non-temporal.
// ---------------------------------------------------------------------------
__global__ __launch_bounds__(256) void pool_kernel(
    const float* __restrict__ x,
    const int* __restrict__ src, const int* __restrict__ dst,
    const float* __restrict__ y,
    float* __restrict__ xnew, int* __restrict__ enew, int* __restrict__ bnew,
    int EPG, int E) {
  __shared__ float yL[NPG], degL[NPG], disL[NPG], zL[NPG];
  __shared__ int keptL[NPG], newidL[NPG], keepL[KNUM];

  const int g = blockIdx.x;
  const int t = threadIdx.x;
  const int base = g * NPG;
  const int ebase = g * EPG;

  if (t < NPG) {
    yL[t] = y[base + t];
    degL[t] = 1.0f;  // self loop
  }
  __syncthreads();

  for (int e = t; e < EPG; e += 256) {
    int r = src[ebase + e] - base;
    atomicAdd(&degL[r], 1.0f);   // ds_add_f32
  }
  __syncthreads();

  if (t < NPG) {
    float d = degL[t];
    float dis = (d > 0.f) ? rsqrtf(d) : 0.f;
    disL[t] = dis;
    zL[t] = dis * dis * yL[t];   // self-loop contribution
  }
  __syncthreads();

  for (int e = t; e < EPG; e += 256) {
    int r = src[ebase + e] - base;
    int c = dst[ebase + e] - base;
    atomicAdd(&zL[c], disL[r] * disL[c] * yL[r]);
  }
  __syncthreads();

  // Stable rank (matches top_k tie-breaking: larger z first, lower index wins ties)
  if (t < NPG) {
    float zi = zL[t];
    int rk = 0;
#pragma unroll 4
    for (int j = 0; j < NPG; ++j) {
      float zj = zL[j];
      rk += (zj > zi) || (zj == zi && j < t);
    }
    keptL[t] = (rk < KNUM) ? 1 : 0;
  }
  __syncthreads();

  if (t < NPG) {
    if (keptL[t]) {
      int loc = 0;
      for (int j = 0; j < t; ++j) loc += keptL[j];
      newidL[t] = g * KNUM + loc;  // exact global cumsum: each graph keeps KNUM
      keepL[loc] = t;
    } else {
      newidL[t] = -1;
    }
  }
  __syncthreads();

  // x_new: coalesced 512B row segments, streaming stores
  for (int idx = t; idx < KNUM * FDIM; idx += 256) {
    int row = idx >> 7;
    int col = idx & (FDIM - 1);
    int node = keepL[row];
    float s = 1.0f / (1.0f + expf(-zL[node]));
    float v = x[(size_t)(base + node) * FDIM + col] * s;
    __builtin_nontemporal_store(v, &xnew[(size_t)(g * KNUM + row) * FDIM + col]);
  }

  if (t < KNUM) __builtin_nontemporal_store(g, &bnew[g * KNUM + t]);

  // edge relabel: dropped edges -> -1 in both rows
  for (int e = t; e < EPG; e += 256) {
    int s0 = src[ebase + e] - base;
    int d0 = dst[ebase + e] - base;
    int ns = newidL[s0];
    int nd = newidL[d0];
    bool keep = (ns >= 0) && (nd >= 0);
    __builtin_nontemporal_store(keep ? ns : -1, &enew[ebase + e]);
    __builtin_nontemporal_store(keep ? nd : -1, &enew[(size_t)E + ebase + e]);
  }
}

// ---------------------------------------------------------------------------
extern "C" void kernel_launch(void* const* d_in, const int* in_sizes, int n_in,
                              void* d_out, int out_size, void* d_ws, size_t ws_size,
                              hipStream_t stream) {
  const float* x     = (const float*)d_in[0];
  const int*   ei    = (const int*)d_in[1];   // (2, E): src row then dst row
  const float* theta = (const float*)d_in[3];

  const int N = in_sizes[2];        // batch vector length == num nodes
  const int E = in_sizes[1] / 2;
  const int Bg = N / NPG;
  const int EPG = E / Bg;

  float* y = (float*)d_ws;          // N floats of scratch

  float* outX = (float*)d_out;
  int*   outE = (int*)(outX + (size_t)Bg * KNUM * FDIM);
  int*   outB = outE + (size_t)2 * E;

  dim3 grid1((N + TILE_ROWS - 1) / TILE_ROWS);
  yproj_wmma_kernel<<<grid1, 128, 0, stream>>>(x, theta, y, N);

  pool_kernel<<<Bg, 256, 0, stream>>>(x, ei, ei + E, y, outX, outE, outB, EPG, E);
}